// RecursivePriorStateSpace_55989193671026
// MI455X (gfx1250) — compile-verified
//
#include <hip/hip_runtime.h>
#include <hip/hip_bf16.h>

typedef __attribute__((ext_vector_type(2))) float v2f;
typedef __attribute__((ext_vector_type(8))) float v8f;

#define HID     128
#define CIN     64
#define PIX     16384      // 128*128 pixels per batch plane
// K-pair-interleaved LDS layouts: rows (2t,2t+1) stored as [pixel][2] pairs.
// Pair stride % 64 == 32 banks -> the two half-wave B-fragment reads hit
// disjoint bank groups; each fragment is one aligned ds_load_b64.
#define PSTR1   544        // K1: 2*256 px + 32 pad floats
#define PSTR3   288        // K3: 2*128 px + 32 pad floats
#define LDSTR2  132        // K2 plane row stride (floats); 16B-aligned rows, <=2-way conflicts

__device__ __forceinline__ float bilinear_prior(const float* __restrict__ prb, int p) {
  // prb: [32,32] plane for this batch; p = pixel index in 128x128, align_corners=True
  int h = p >> 7, w = p & 127;
  const float s = 31.0f / 127.0f;
  float sy = h * s, sx = w * s;
  int y0 = (int)sy, x0 = (int)sx;
  int y1 = (y0 + 1 > 31) ? 31 : y0 + 1;
  int x1 = (x0 + 1 > 31) ? 31 : x0 + 1;
  float fy = sy - (float)y0, fx = sx - (float)x0;
  float p00 = prb[(y0 << 5) + x0], p01 = prb[(y0 << 5) + x1];
  float p10 = prb[(y1 << 5) + x0], p11 = prb[(y1 << 5) + x1];
  float v = (p00 * (1.f - fx) + p01 * fx) * (1.f - fy)
          + (p10 * (1.f - fx) + p11 * fx) * fy;
  return fminf(fmaxf(v, -1.f), 1.f);
}

// ---------------------------------------------------------------------------
// K1: x_proj[b, hid, p] = W_in @ (x + alpha*pr) + b_in     (f32 WMMA 16x16x4)
// Block: 256 px of one batch, all 128 hid rows. 8 waves = 8 hid tiles of 16.
// ---------------------------------------------------------------------------
__global__ __launch_bounds__(256) void k1_inproj(
    const float* __restrict__ x, const float* __restrict__ prior,
    const float* __restrict__ Win, const float* __restrict__ bin,
    const float* __restrict__ alpha_p, float* __restrict__ xproj)
{
  extern __shared__ float lds[];                 // 32 K-pairs x PSTR1 + 256 pr
  float* pr_s = lds + 32 * PSTR1;

  const int tid   = threadIdx.x;
  const int pb    = blockIdx.x;                  // 512 blocks
  const int b     = pb >> 6;
  const int pbase = (pb & 63) << 8;              // 256-pixel block
  const float alpha = *alpha_p;

  const float* xb  = x + ((size_t)b << 20);      // b*64*16384
  const float* prb = prior + b * 1024;           // b*32*32

  // stage pr for the 256 pixels (computed once, shared by all 64 channels)
  pr_s[tid] = bilinear_prior(prb, pbase + tid);
  __syncthreads();

  // load F_mod block pair-interleaved: 32 channel pairs x 64 pixel quads
  for (int i = tid; i < 2048; i += 256) {
    int t = i >> 6, q = i & 63;                  // t: channel pair, q: pixel quad
    int po = q << 2;
    float4 r0 = *(const float4*)(xb + ((size_t)(2 * t)     << 14) + pbase + po);
    float4 r1 = *(const float4*)(xb + ((size_t)(2 * t + 1) << 14) + pbase + po);
    float pr0 = alpha * pr_s[po + 0], pr1 = alpha * pr_s[po + 1];
    float pr2 = alpha * pr_s[po + 2], pr3 = alpha * pr_s[po + 3];
    float* dst = &lds[t * PSTR1 + (po << 1)];
    v2f e0 = { r0.x + pr0, r1.x + pr0 };
    v2f e1 = { r0.y + pr1, r1.y + pr1 };
    v2f e2 = { r0.z + pr2, r1.z + pr2 };
    v2f e3 = { r0.w + pr3, r1.w + pr3 };
    *(v2f*)(dst + 0) = e0;
    *(v2f*)(dst + 2) = e1;
    *(v2f*)(dst + 4) = e2;
    *(v2f*)(dst + 6) = e3;
  }
  __syncthreads();

  const int wv   = tid >> 5;
  const int lane = tid & 31;
  const int half = lane >> 4;                    // 0: lanes 0-15, 1: lanes 16-31
  const int l15  = lane & 15;
  const int m0   = wv << 4;                      // hid tile base

  // Preload A fragments (W_in rows m0..m0+15), 16 K-chunks of 4
  v2f a_frag[16];
#pragma unroll
  for (int kc = 0; kc < 16; ++kc) {
    const float* wr = Win + (m0 + l15) * CIN + (kc << 2) + (half << 1);
    a_frag[kc].x = wr[0];
    a_frag[kc].y = wr[1];
  }
  float binv[8];
#pragma unroll
  for (int j = 0; j < 8; ++j) binv[j] = bin[m0 + j + (half << 3)];

  const float* bbase = &lds[half * PSTR1 + (l15 << 1)];
  float* outp_base = xproj + ((size_t)b << 21) + pbase;   // b*128*16384

  for (int nc = 0; nc < 16; ++nc) {              // 16 pixel tiles of 16
    v8f acc = {};
#pragma unroll
    for (int kc = 0; kc < 16; ++kc) {
      // one aligned 8-byte LDS read per WMMA (rows 4kc+2half, 4kc+2half+1)
      v2f bf = *(const v2f*)(bbase + (kc << 1) * PSTR1 + (nc << 5));
      acc = __builtin_amdgcn_wmma_f32_16x16x4_f32(
          false, a_frag[kc], false, bf, (short)0, acc, false, false);
    }
    float* outp = outp_base + (nc << 4) + l15;
#pragma unroll
    for (int j = 0; j < 8; ++j) {
      int m = m0 + j + (half << 3);
      outp[(size_t)m << 14] = acc[j] + binv[j];
    }
  }
}

// ---------------------------------------------------------------------------
// K2: in-place scans over x_proj: S = hscan(x_proj) + vscan(x_proj)
// Block: one batch, 2 hid channels; both 128x128 planes resident in LDS.
// ---------------------------------------------------------------------------
__global__ __launch_bounds__(256) void k2_scan(
    const float* __restrict__ Av, const float* __restrict__ Bv,
    float* __restrict__ xproj)
{
  extern __shared__ float lds[];                 // 2 in-planes + 2 sum-planes, stride 132
  const int PLANE = 128 * LDSTR2;
  float* sum0 = lds + 2 * PLANE;

  const int tid = threadIdx.x;
  const int blk = blockIdx.x;                    // 512 blocks
  const int b   = blk >> 6;
  const int c0  = (blk & 63) << 1;

  float* gp = xproj + ((size_t)b << 21) + ((size_t)c0 << 14);

  // load 2 planes (8192 float4s)
  for (int i = tid; i < 8192; i += 256) {
    int p = i >> 12, r = (i >> 5) & 127, q = i & 31;
    float4 v = *(const float4*)(gp + ((size_t)p << 14) + (r << 7) + (q << 2));
    *(float4*)(&lds[p * PLANE + r * LDSTR2 + (q << 2)]) = v;
  }
  __syncthreads();

  float Ad0 = tanhf(Av[c0]),     Ad1 = tanhf(Av[c0 + 1]);
  float Bd0 = Bv[c0],            Bd1 = Bv[c0 + 1];

  {   // horizontal scan: one thread per (plane,row)
    int p = tid >> 7, r = tid & 127;
    float Ad = p ? Ad1 : Ad0, Bd = p ? Bd1 : Bd0;
    const float* row = &lds[p * PLANE + r * LDSTR2];
    float* srow = &sum0[p * PLANE + r * LDSTR2];
    float s = 0.f;
#pragma unroll 8
    for (int w = 0; w < 128; ++w) { s = Ad * s + Bd * row[w]; srow[w] = s; }
  }
  __syncthreads();

  {   // vertical scan: one thread per (plane,col), accumulate into sum
    int p = tid >> 7, col = tid & 127;
    float Ad = p ? Ad1 : Ad0, Bd = p ? Bd1 : Bd0;
    const float* base = &lds[p * PLANE + col];
    float* sb = &sum0[p * PLANE + col];
    float s = 0.f;
#pragma unroll 8
    for (int h = 0; h < 128; ++h) {
      s = Ad * s + Bd * base[h * LDSTR2];
      sb[h * LDSTR2] += s;
    }
  }
  __syncthreads();

  // store back in place
  for (int i = tid; i < 8192; i += 256) {
    int p = i >> 12, r = (i >> 5) & 127, q = i & 31;
    *(float4*)(gp + ((size_t)p << 14) + (r << 7) + (q << 2)) =
        *(float4*)(&sum0[p * PLANE + r * LDSTR2 + (q << 2)]);
  }
}

// ---------------------------------------------------------------------------
// K3: out = (x + alpha*pr) + gamma * (W_out @ S + b_out)   (f32 WMMA 16x16x4)
// Block: 128 px, all 64 out channels. 8 waves: 4 M-tiles x 2 pixel halves.
// ---------------------------------------------------------------------------
__global__ __launch_bounds__(256) void k3_outproj(
    const float* __restrict__ x, const float* __restrict__ prior,
    const float* __restrict__ Wout, const float* __restrict__ bout,
    const float* __restrict__ alpha_p, const float* __restrict__ gamma_p,
    const float* __restrict__ S, float* __restrict__ out)
{
  extern __shared__ float lds[];                 // 64 K-pairs x PSTR3 + 128 pr
  float* pr_s = lds + 64 * PSTR3;

  const int tid   = threadIdx.x;
  const int pb    = blockIdx.x;                  // 1024 blocks
  const int b     = pb >> 7;
  const int pbase = (pb & 127) << 7;             // 128-pixel block
  const float alpha = *alpha_p, gamma = *gamma_p;

  const float* prb = prior + b * 1024;
  if (tid < 128) pr_s[tid] = bilinear_prior(prb, pbase + tid);

  // load S block pair-interleaved: 64 hid pairs x 32 pixel quads
  const float* Sb = S + ((size_t)b << 21) + pbase;
  for (int i = tid; i < 2048; i += 256) {
    int t = i >> 5, q = i & 31;
    int po = q << 2;
    float4 r0 = *(const float4*)(Sb + ((size_t)(2 * t)     << 14) + po);
    float4 r1 = *(const float4*)(Sb + ((size_t)(2 * t + 1) << 14) + po);
    float* dst = &lds[t * PSTR3 + (po << 1)];
    v2f e0 = { r0.x, r1.x };
    v2f e1 = { r0.y, r1.y };
    v2f e2 = { r0.z, r1.z };
    v2f e3 = { r0.w, r1.w };
    *(v2f*)(dst + 0) = e0;
    *(v2f*)(dst + 2) = e1;
    *(v2f*)(dst + 4) = e2;
    *(v2f*)(dst + 6) = e3;
  }
  __syncthreads();

  const int wv   = tid >> 5;
  const int lane = tid & 31;
  const int half = lane >> 4;
  const int l15  = lane & 15;
  const int m0     = (wv & 3) << 4;              // out-channel tile base
  const int ncbase = (wv >> 2) << 2;             // pixel-tile group (0 or 4)

  // Preload W_out A fragments: 32 K-chunks of 4
  v2f a_frag[32];
#pragma unroll
  for (int kc = 0; kc < 32; ++kc) {
    const float* wr = Wout + (m0 + l15) * HID + (kc << 2) + (half << 1);
    a_frag[kc].x = wr[0];
    a_frag[kc].y = wr[1];
  }
  float boutv[8];
#pragma unroll
  for (int j = 0; j < 8; ++j) boutv[j] = bout[m0 + j + (half << 3)];

  const float* bbase = &lds[half * PSTR3 + (l15 << 1)];
  const float* xb = x + ((size_t)b << 20) + pbase;
  float* ob = out + ((size_t)b << 20) + pbase;

  for (int nci = 0; nci < 4; ++nci) {
    int nc = ncbase + nci;
    v8f acc = {};
#pragma unroll
    for (int kc = 0; kc < 32; ++kc) {
      // one aligned 8-byte LDS read per WMMA (rows 4kc+2half, 4kc+2half+1)
      v2f bf = *(const v2f*)(bbase + (kc << 1) * PSTR3 + (nc << 5));
      acc = __builtin_amdgcn_wmma_f32_16x16x4_f32(
          false, a_frag[kc], false, bf, (short)0, acc, false, false);
    }
    int po  = (nc << 4) + l15;
    float prv = pr_s[po];
#pragma unroll
    for (int j = 0; j < 8; ++j) {
      int m = m0 + j + (half << 3);
      float fmod = xb[((size_t)m << 14) + po] + alpha * prv;
      ob[((size_t)m << 14) + po] = fmod + gamma * (acc[j] + boutv[j]);
    }
  }
}

// ---------------------------------------------------------------------------
extern "C" void kernel_launch(void* const* d_in, const int* in_sizes, int n_in,
                              void* d_out, int out_size, void* d_ws, size_t ws_size,
                              hipStream_t stream) {
  const float* x      = (const float*)d_in[0];   // [8,64,128,128]
  const float* prior  = (const float*)d_in[1];   // [8,1,32,32]
  const float* Win    = (const float*)d_in[2];   // [128,64]
  const float* bin    = (const float*)d_in[3];   // [128]
  const float* Av     = (const float*)d_in[4];   // [128]
  const float* Bv     = (const float*)d_in[5];   // [128]
  const float* alpha  = (const float*)d_in[6];   // scalar
  const float* gamma  = (const float*)d_in[7];   // scalar
  const float* Wout   = (const float*)d_in[8];   // [64,128]
  const float* bout   = (const float*)d_in[9];   // [64]
  float* out = (float*)d_out;

  float* xproj = (float*)d_ws;                   // [8,128,128,128] = 64 MiB, scanned in place

  const size_t lds1 = (size_t)(32 * PSTR1 + 256) * sizeof(float);       //  70,656 B
  const size_t lds2 = (size_t)(4 * 128 * LDSTR2) * sizeof(float);       // 270,336 B (of 320 KB/WGP)
  const size_t lds3 = (size_t)(64 * PSTR3 + 128) * sizeof(float);       //  74,240 B

  k1_inproj <<<dim3(512),  dim3(256), lds1, stream>>>(x, prior, Win, bin, alpha, xproj);
  k2_scan   <<<dim3(512),  dim3(256), lds2, stream>>>(Av, Bv, xproj);
  k3_outproj<<<dim3(1024), dim3(256), lds3, stream>>>(x, prior, Wout, bout, alpha, gamma,
                                                      xproj, out);
}